// BaseMessageModule_86706799772382
// MI455X (gfx1250) — compile-verified
//
#include <hip/hip_runtime.h>
#include <stdint.h>

// ---------------------------------------------------------------------------
// BaseMessageModule scatter/gather kernel for MI455X (gfx1250, wave32).
//
// Bandwidth-bound: f_ij_cutoff (512 MB) is the only HBM stream; features
// (51 MB) and radial output (51 MB) are L2-resident. f_ij tiles are staged
// into LDS by the Tensor Data Mover (tensor_load_to_lds, TENSORcnt), double
// buffered so the DMA of chunk c+1 overlaps processing of chunk c.
// ---------------------------------------------------------------------------

#define NFEAT 128
#define CHUNK 64                    // edges per TDM tile (64 x 128 f32 = 32 KB)
#define WAVES_PER_BLOCK 8
#define ROWS_PER_WAVE (CHUNK / WAVES_PER_BLOCK)

typedef unsigned int u32x4 __attribute__((ext_vector_type(4)));
typedef int          i32x4 __attribute__((ext_vector_type(4)));
typedef int          i32x8 __attribute__((ext_vector_type(8)));

// Issue a 2D TDM load: tile_rows x 128 f32 tile of a row-major [*, 128]
// tensor -> LDS at lds_byte_addr.  D# layout per CDNA5 ISA
// §8.3 (group 0) / §8.4 (group 1).  Groups 2/3 (and the trailing octet of the
// clang-23 builtin) are unused for 2D tensors -> zero-filled.
__device__ __forceinline__ void tdm_load_tile_2d(uint32_t lds_byte_addr,
                                                 const float* gsrc,
                                                 uint32_t tile_rows,
                                                 uint32_t rows_left) {
  const uint64_t ga = (uint64_t)(uintptr_t)gsrc;
  u32x4 g0;
  g0[0] = 1u;                                   // count=1 valid descriptor
  g0[1] = lds_byte_addr;                        // lds_addr [63:32]
  g0[2] = (uint32_t)ga;                         // global_addr lo
  g0[3] = ((uint32_t)(ga >> 32) & 0x01FFFFFFu)  // global_addr [56:32]
          | (2u << 30);                         // type = 2 ("image")
  i32x8 g1;
  g1[0] = (int)(2u << 16);                      // data_size = 2 (4 bytes)
  g1[1] = (int)(((uint32_t)NFEAT & 0xFFFFu) << 16);  // tensor_dim0 lo16
  g1[2] = (int)(((uint32_t)NFEAT >> 16)              // tensor_dim0 hi16
                | ((rows_left & 0xFFFFu) << 16));    // tensor_dim1 lo16
  g1[3] = (int)((rows_left >> 16)                    // tensor_dim1 hi16
                | ((uint32_t)NFEAT << 16));          // tile_dim0 = 128
  g1[4] = (int)tile_rows;                       // tile_dim1 (tile_dim2 = 0)
  g1[5] = (int)NFEAT;                           // tensor_dim0_stride lo32 = 128
  g1[6] = 0;                                    // stride0 hi / stride1 lo
  g1[7] = 0;                                    // stride1 hi
  const i32x4 z4 = {0, 0, 0, 0};
  const i32x8 z8 = {0, 0, 0, 0, 0, 0, 0, 0};
#if __has_builtin(__builtin_amdgcn_tensor_load_to_lds)
  // amdgpu-toolchain (clang-23) arity: (g0, g1, g2, g3, g4, cpol)
  __builtin_amdgcn_tensor_load_to_lds(g0, g1, z4, z4, z8, 0);
#else
  (void)z4; (void)z8;
#endif
}

__device__ __forceinline__ void wait_tensorcnt0() {
#if __has_builtin(__builtin_amdgcn_s_wait_tensorcnt)
  __builtin_amdgcn_s_wait_tensorcnt(0);
#else
  asm volatile("s_wait_tensorcnt 0x0" ::: "memory");
#endif
}
__device__ __forceinline__ void wait_tensorcnt1() {
#if __has_builtin(__builtin_amdgcn_s_wait_tensorcnt)
  __builtin_amdgcn_s_wait_tensorcnt(1);
#else
  asm volatile("s_wait_tensorcnt 0x1" ::: "memory");
#endif
}

__device__ __forceinline__ void atomic_add_f32(float* p, float v) {
  // Relaxed agent-scope fp32 add -> global_atomic_add_f32 (no return).
  __hip_atomic_fetch_add(p, v, __ATOMIC_RELAXED, __HIP_MEMORY_SCOPE_AGENT);
}

// ---------------------------------------------------------------------------
__global__ void zero_kernel(float* __restrict__ p, long n) {
  const long i = (long)blockIdx.x * blockDim.x + threadIdx.x;
  const long stride = (long)gridDim.x * blockDim.x;
  const long n4 = n >> 2;
  float4* p4 = (float4*)p;
  for (long k = i; k < n4; k += stride)
    p4[k] = make_float4(0.f, 0.f, 0.f, 0.f);
  for (long k = (n4 << 2) + i; k < n; k += stride)
    p[k] = 0.f;
}

// ---------------------------------------------------------------------------
// One wave per edge: 32 lanes x float4 = 128 features.
__global__ __launch_bounds__(WAVES_PER_BLOCK * 32)
void edge_kernel(const float* __restrict__ features,   // [N,128]
                 const int*   __restrict__ idx_j,      // [E]
                 const float* __restrict__ f_ij,       // [E,128]
                 const float* __restrict__ r_ij,       // [E,3]
                 float* __restrict__ radial,           // [N,128] accum
                 float* __restrict__ vec,              // [N,3]  accum
                 long E, long nchunks) {
  __shared__ float tile[2][CHUNK * NFEAT];             // 2 x 32 KB
  const int lane = threadIdx.x & 31;
  const int wave = threadIdx.x >> 5;

  long c = blockIdx.x;
  const long step = gridDim.x;

  if (c < nchunks && wave == 0) {
    const long e0 = c * CHUNK;
    const long rem = E - e0;
    const uint32_t rows = (uint32_t)(rem < (long)CHUNK ? rem : (long)CHUNK);
    tdm_load_tile_2d((uint32_t)(uintptr_t)&tile[0][0],
                     f_ij + (size_t)e0 * NFEAT, rows, (uint32_t)rem);
  }

  int it = 0;
  for (; c < nchunks; c += step, ++it) {
    const int buf = it & 1;
    const long cn = c + step;
    if (wave == 0) {
      if (cn < nchunks) {
        const long e0n = cn * CHUNK;
        const long remn = E - e0n;
        const uint32_t rows = (uint32_t)(remn < (long)CHUNK ? remn : (long)CHUNK);
        tdm_load_tile_2d((uint32_t)(uintptr_t)&tile[buf ^ 1][0],
                         f_ij + (size_t)e0n * NFEAT, rows, (uint32_t)remn);
        wait_tensorcnt1();            // chunk c resident; c+1 still in flight
      } else {
        wait_tensorcnt0();
      }
    }
    __syncthreads();                  // publish LDS tile to all waves

    const long e_base = c * CHUNK;
#pragma unroll
    for (int k = 0; k < ROWS_PER_WAVE; ++k) {
      const int row = wave * ROWS_PER_WAVE + k;
      const long e = e_base + row;
      if (e >= E) break;              // wave-uniform

      const int j = idx_j[e];
      const float rx = r_ij[3 * e + 0];
      const float ry = r_ij[3 * e + 1];
      const float rz = r_ij[3 * e + 2];
      const float inv = rsqrtf(rx * rx + ry * ry + rz * rz);

      const float4 fv = ((const float4*)&tile[buf][row * NFEAT])[lane]; // ds_load_b128
      const float4 gv = ((const float4*)(features + (size_t)j * NFEAT))[lane]; // L2 gather

      float4 p;
      p.x = fv.x * gv.x; p.y = fv.y * gv.y;
      p.z = fv.z * gv.z; p.w = fv.w * gv.w;

      float* dst = radial + (size_t)j * NFEAT + lane * 4;
      atomic_add_f32(dst + 0, p.x);
      atomic_add_f32(dst + 1, p.y);
      atomic_add_f32(dst + 2, p.z);
      atomic_add_f32(dst + 3, p.w);

      // wave32 reduction of sum(proto[e])
      float s = (p.x + p.y) + (p.z + p.w);
#pragma unroll
      for (int o = 16; o > 0; o >>= 1) s += __shfl_xor(s, o, 32);

      if (lane < 3) {
        const float u = (lane == 0 ? rx : (lane == 1 ? ry : rz)) * inv;
        atomic_add_f32(vec + 3 * (size_t)j + lane, u * s);
      }
    }
    __syncthreads();                  // all waves done reading tile[buf]
  }
}

// ---------------------------------------------------------------------------
__global__ void norm_kernel(const float* __restrict__ vec,  // [N,3]
                            float* __restrict__ out,        // [N]
                            int n) {
  const int i = blockIdx.x * blockDim.x + threadIdx.x;
  if (i < n) {
    const float x = vec[3 * i + 0];
    const float y = vec[3 * i + 1];
    const float z = vec[3 * i + 2];
    out[i] = sqrtf(x * x + y * y + z * z);
  }
}

// ---------------------------------------------------------------------------
extern "C" void kernel_launch(void* const* d_in, const int* in_sizes, int n_in,
                              void* d_out, int out_size, void* d_ws, size_t ws_size,
                              hipStream_t stream) {
  (void)n_in; (void)out_size; (void)ws_size;

  const float* features = (const float*)d_in[0];   // [N,128] f32
  const int*   pairlist = (const int*)  d_in[1];   // [2,E]   int
  const float* f_ij     = (const float*)d_in[2];   // [E,128] f32
  const float* r_ij     = (const float*)d_in[3];   // [E,3]   f32

  const long NF = (long)in_sizes[0];               // N * 128
  const long N  = NF / NFEAT;
  const long E  = (long)in_sizes[1] / 2;
  const int* idx_j = pairlist + E;                 // row 1 of pairlist

  float* radial    = (float*)d_out;                // [N,128]
  float* norms_out = radial + NF;                  // [N]
  float* vec       = (float*)d_ws;                 // [N,3] accumulator

  // Zero accumulators every call (harness poisons buffers; we accumulate).
  zero_kernel<<<2048, 256, 0, stream>>>(radial, NF);
  zero_kernel<<<512, 256, 0, stream>>>(vec, 3 * N);

  const long nchunks = (E + CHUNK - 1) / CHUNK;
  const int blocks = (int)(nchunks < 2048 ? nchunks : 2048);
  edge_kernel<<<blocks, WAVES_PER_BLOCK * 32, 0, stream>>>(
      features, idx_j, f_ij, r_ij, radial, vec, E, nchunks);

  norm_kernel<<<(int)((N + 255) / 256), 256, 0, stream>>>(vec, norms_out, (int)N);
}